// InversePreEmphasis_28707561406921
// MI455X (gfx1250) — compile-verified
//
#include <hip/hip_runtime.h>
#include <stdint.h>

// y[b,t] = h_t,  h_t = tanh(x[b,t] + 0.97*h_{t-1}),  h_{-1}=0.
// Contraction-based segment parallelism: |d h_t / d h_{t-1}| <= 0.97, so a
// 1024-step warm-up from h=0 converges to the true trajectory to ~3e-14,
// far below fp32 epsilon. 64 segments x 256 rows = 16384 independent chains.

typedef float v4f __attribute__((ext_vector_type(4)));

constexpr int   kB     = 256;       // rows (== blockDim.x)
constexpr int   kT     = 131072;    // timesteps
constexpr int   kSeg   = 2048;      // output timesteps per segment
constexpr int   kWarm  = 1024;      // warm-up steps (0.97^1024 ~ 3e-14)
constexpr float kCoef  = 0.97f;
constexpr int   kC     = 64;        // timesteps staged per chunk per thread
constexpr int   kQ     = kC / 4;    // float4s per chunk = 16
constexpr int   kNBuf  = 4;         // LDS buffers (reuse distance > depth)
constexpr int   kDepth = 3;         // chunks in flight: ~3*640cy latency cover
constexpr int   kStrF  = kC + 4;    // 68 floats / 272 B per-thread LDS stride
constexpr int   kBufF  = kB * kStrF;

__device__ __forceinline__ float hw_tanh(float a) {
#if __has_builtin(__builtin_amdgcn_tanhf)
  return __builtin_amdgcn_tanhf(a);
#elif __has_builtin(__builtin_amdgcn_tanh_f32)
  return __builtin_amdgcn_tanh_f32(a);
#else
  float r;
  // v_tanh_f32 is a TRANS op: dependent consumer needs 1 delay slot.
  asm("v_tanh_f32 %0, %1\n\ts_delay_alu instid0(TRANS32_DEP_1)"
      : "=v"(r)
      : "v"(a));
  return r;
#endif
}

// Stage one chunk (kC floats = kQ x 16B per lane) global -> LDS.
// INST_OFFSET is added to BOTH the LDS and global address (ISA 08 §4.4),
// which matches our layout exactly: +16B in global == +16B in the LDS slot.
__device__ __forceinline__ void async_chunk(unsigned lds_base, const float* g) {
#pragma unroll
  for (int k = 0; k < kQ; ++k) {
    asm volatile("global_load_async_to_lds_b128 %0, %1, off offset:%2"
                 :: "v"(lds_base), "v"(g), "i"(16 * k)
                 : "memory");
  }
}

extern "C" __global__ void __launch_bounds__(kB)
ipe_scan_kernel(const float* __restrict__ x, float* __restrict__ y) {
  extern __shared__ float smem[];

  const int tid = threadIdx.x;   // row
  const int seg = blockIdx.x;    // time segment

  const int warm_ch = (seg == 0) ? 0 : (kWarm / kC);   // 0 or 16
  const int nch     = kSeg / kC + warm_ch;             // 32 or 48 chunks
  const int t0      = seg * kSeg - warm_ch * kC;       // first staged timestep

  const float* gx = x + (size_t)tid * kT + t0;
  float*       gy = y + (size_t)tid * kT + t0;

  // Per-buffer LDS byte addresses for this thread's slot. Generic->int
  // truncation yields the LDS byte offset (flat LDS addr low 32 bits).
  unsigned lbase[kNBuf];
#pragma unroll
  for (int b = 0; b < kNBuf; ++b)
    lbase[b] = (unsigned)(uintptr_t)&smem[b * kBufF + tid * kStrF];

  // Prologue: fill the pipe (nch >= 32 > kDepth always).
#pragma unroll
  for (int c = 0; c < kDepth; ++c)
    async_chunk(lbase[c & (kNBuf - 1)], gx + c * kC);

  float h = 0.0f;
  for (int c = 0; c < nch; ++c) {
    // Wait until chunk c's kQ async ops are complete. Completion is in-order:
    // allowed outstanding = (#chunk-ops issued after chunk c). Tail-safe.
    const int after = nch - 1 - c;
    if (after >= kDepth - 1) {
      asm volatile("s_wait_asynccnt %0" :: "i"((kDepth - 1) * kQ) : "memory");
    } else if (after == 1) {
      asm volatile("s_wait_asynccnt %0" :: "i"(kQ) : "memory");
    } else {
      asm volatile("s_wait_asynccnt 0" ::: "memory");
    }

    // Restart the copy engine ASAP: buffer (c+kDepth)&3 is disjoint from the
    // live buffers c, c+1, c+2 since kNBuf > kDepth.
    if (c + kDepth < nch)
      async_chunk(lbase[(c + kDepth) & (kNBuf - 1)], gx + (c + kDepth) * kC);

    // Pull the whole chunk LDS -> VGPRs up front (ds_load_b128 x16), so the
    // serial fma+tanh chain never waits on memory.
    const v4f* sb =
        (const v4f*)&smem[(c & (kNBuf - 1)) * kBufF + tid * kStrF];
    v4f xv[kQ];
#pragma unroll
    for (int k = 0; k < kQ; ++k) xv[k] = sb[k];

    const bool emit = (c >= warm_ch);  // uniform per block
#pragma unroll
    for (int k = 0; k < kQ; ++k) {
      const v4f v = xv[k];
      v4f o;
      h = hw_tanh(__builtin_fmaf(kCoef, h, v.x)); o.x = h;
      h = hw_tanh(__builtin_fmaf(kCoef, h, v.y)); o.y = h;
      h = hw_tanh(__builtin_fmaf(kCoef, h, v.z)); o.z = h;
      h = hw_tanh(__builtin_fmaf(kCoef, h, v.w)); o.w = h;
      if (emit)
        __builtin_nontemporal_store(o, (v4f*)(gy + c * kC + 4 * k));
    }
  }
}

extern "C" void kernel_launch(void* const* d_in, const int* in_sizes, int n_in,
                              void* d_out, int out_size, void* d_ws, size_t ws_size,
                              hipStream_t stream) {
  (void)in_sizes; (void)n_in; (void)out_size; (void)d_ws; (void)ws_size;
  const float* x = (const float*)d_in[0];
  float*       y = (float*)d_out;

  dim3 grid(kT / kSeg);   // 64 segments
  dim3 block(kB);         // 256 threads = 8 wave32
  size_t shmem = (size_t)kNBuf * kBufF * sizeof(float);  // 272 KB of 320 KB LDS
  ipe_scan_kernel<<<grid, block, shmem, stream>>>(x, y);
}